// GINConv_31121333027433
// MI455X (gfx1250) — compile-verified
//
#include <hip/hip_runtime.h>

typedef __attribute__((ext_vector_type(2))) float v2f;
typedef __attribute__((ext_vector_type(8))) float v8f;

#define N_NODES 50000
#define N_EDGES 800000
#define D 128
#define LDS_PITCH 132   // 128 + 4 pad: breaks stride-128 bank conflicts on A-frag loads

// ---------------------------------------------------------------------------
// Kernel 1: rst = (1 + eps) * feat   (float4 vectorized, 6.4M elements)
// ---------------------------------------------------------------------------
__global__ void gin_init_rst(const float* __restrict__ feat,
                             const float* __restrict__ eps,
                             float* __restrict__ rst) {
    int i = blockIdx.x * blockDim.x + threadIdx.x;   // float4 index
    float s = 1.0f + eps[0];
    const float4* f4 = (const float4*)feat;
    float4* r4 = (float4*)rst;
    float4 v = f4[i];
    v.x *= s; v.y *= s; v.z *= s; v.w *= s;
    r4[i] = v;
}

// ---------------------------------------------------------------------------
// Kernel 2: edge scatter-sum. One wave32 per edge: each lane gathers a float4
// of feat[src] (512B coalesced per wave) and fires 4 native f32 atomic adds
// into rst[dst]. feat/rst both fit in the 192MB L2, so this stays on-chip.
// ---------------------------------------------------------------------------
__global__ void gin_scatter(const float* __restrict__ feat,
                            const int* __restrict__ esrc,
                            const int* __restrict__ edst,
                            float* __restrict__ rst) {
    int wave = (blockIdx.x * blockDim.x + threadIdx.x) >> 5;
    int lane = threadIdx.x & 31;
    if (wave >= N_EDGES) return;
    int s = __builtin_amdgcn_readfirstlane(esrc[wave]);
    int d = __builtin_amdgcn_readfirstlane(edst[wave]);
    const float4* fs = (const float4*)(feat + (size_t)s * D);
    float4 v = fs[lane];
    float* o = rst + (size_t)d * D + lane * 4;
    unsafeAtomicAdd(o + 0, v.x);
    unsafeAtomicAdd(o + 1, v.y);
    unsafeAtomicAdd(o + 2, v.z);
    unsafeAtomicAdd(o + 3, v.w);
}

// ---------------------------------------------------------------------------
// Kernel 3: fused 2-layer MLP via V_WMMA_F32_16X16X4_F32.
// Block = 256 threads (8 waves) owns 16 rows; wave w owns output cols
// [16w, 16w+16). Layer-1 A tile staged once in LDS (shared by all 8 waves),
// layer-1 output (post-ReLU) kept in LDS as layer-2's A matrix.
// ---------------------------------------------------------------------------
__global__ void __launch_bounds__(256)
gin_mlp(const float* __restrict__ rst,
        const float* __restrict__ W1, const float* __restrict__ b1,
        const float* __restrict__ W2, const float* __restrict__ b2,
        float* __restrict__ out) {
    __shared__ float hA[16 * LDS_PITCH];   // rst tile   (layer-1 A)
    __shared__ float hH[16 * LDS_PITCH];   // hidden tile (layer-2 A)

    const int tid  = threadIdx.x;
    const int wave = tid >> 5;
    const int lane = tid & 31;
    const int l15  = lane & 15;
    const int hi   = lane >> 4;        // half-wave select
    const int koff = hi * 2;           // A/B fragment K offset per §7.12.2
    const int m0   = blockIdx.x * 16;
    const int n0   = wave * 16;

    // --- cooperative load: 16x128 rst tile -> LDS (8 floats per thread) ---
    {
        int r = tid >> 4;              // row 0..15
        int c = (tid & 15) * 8;        // col start
        const float4* src = (const float4*)(rst + (size_t)(m0 + r) * D + c);
        float4 a = src[0], b = src[1];
        float* dst = hA + r * LDS_PITCH + c;
        dst[0] = a.x; dst[1] = a.y; dst[2] = a.z; dst[3] = a.w;
        dst[4] = b.x; dst[5] = b.y; dst[6] = b.z; dst[7] = b.w;
    }
    __syncthreads();

    // --- layer 1: acc = rst_tile @ W1 + b1 -------------------------------
    v8f acc;
    {
        float bb = b1[n0 + l15];       // C/D layout: N = n0 + (lane&15) for all rows
        #pragma unroll
        for (int r = 0; r < 8; ++r) acc[r] = bb;
    }
    #pragma unroll
    for (int k = 0; k < D; k += 4) {
        v2f a = *(const v2f*)(hA + l15 * LDS_PITCH + k + koff);
        v2f b;
        b.x = W1[(size_t)(k + koff)     * D + n0 + l15];
        b.y = W1[(size_t)(k + koff + 1) * D + n0 + l15];
        acc = __builtin_amdgcn_wmma_f32_16x16x4_f32(
            /*neg_a=*/false, a, /*neg_b=*/false, b,
            /*c_mod=*/(short)0, acc, /*reuse_a=*/false, /*reuse_b=*/false);
    }
    // ReLU -> hidden tile in LDS
    #pragma unroll
    for (int r = 0; r < 8; ++r) {
        int m = r + hi * 8;
        float v = acc[r] > 0.0f ? acc[r] : 0.0f;
        hH[m * LDS_PITCH + n0 + l15] = v;
    }
    __syncthreads();

    // --- layer 2: out = relu(h_tile @ W2 + b2) ---------------------------
    v8f acc2;
    {
        float bb = b2[n0 + l15];
        #pragma unroll
        for (int r = 0; r < 8; ++r) acc2[r] = bb;
    }
    #pragma unroll
    for (int k = 0; k < D; k += 4) {
        v2f a = *(const v2f*)(hH + l15 * LDS_PITCH + k + koff);
        v2f b;
        b.x = W2[(size_t)(k + koff)     * D + n0 + l15];
        b.y = W2[(size_t)(k + koff + 1) * D + n0 + l15];
        acc2 = __builtin_amdgcn_wmma_f32_16x16x4_f32(
            false, a, false, b, (short)0, acc2, false, false);
    }
    #pragma unroll
    for (int r = 0; r < 8; ++r) {
        int m = r + hi * 8;
        float v = acc2[r] > 0.0f ? acc2[r] : 0.0f;
        out[(size_t)(m0 + m) * D + n0 + l15] = v;
    }
}

// ---------------------------------------------------------------------------
extern "C" void kernel_launch(void* const* d_in, const int* in_sizes, int n_in,
                              void* d_out, int out_size, void* d_ws, size_t ws_size,
                              hipStream_t stream) {
    const float* feat = (const float*)d_in[0];
    const float* eps  = (const float*)d_in[1];
    const float* W1   = (const float*)d_in[2];
    const float* b1   = (const float*)d_in[3];
    const float* W2   = (const float*)d_in[4];
    const float* b2   = (const float*)d_in[5];
    const int*   esrc = (const int*)d_in[6];
    const int*   edst = (const int*)d_in[7];
    float* out = (float*)d_out;
    float* rst = (float*)d_ws;                 // 50000*128 fp32 = 25.6 MB scratch

    // rst = (1+eps)*feat
    gin_init_rst<<<(N_NODES * D / 4) / 256, 256, 0, stream>>>(feat, eps, rst);
    // rst += segment_sum(feat[src] -> dst)
    gin_scatter<<<(N_EDGES * 32) / 256, 256, 0, stream>>>(feat, esrc, edst, rst);
    // out = relu(relu(rst@W1+b1)@W2+b2)
    gin_mlp<<<N_NODES / 16, 256, 0, stream>>>(rst, W1, b1, W2, b2, out);
}